// GNNNet_678604833376
// MI455X (gfx1250) — compile-verified
//
#include <hip/hip_runtime.h>
#include <hip/hip_bf16.h>

typedef float v2f __attribute__((ext_vector_type(2)));
typedef float v8f __attribute__((ext_vector_type(8)));

#define D_FEAT 128
#define BATCH 2

// ---- Kernel 1: accumulate weighted in-degree over edges ----
__global__ void deg_accum(const int* __restrict__ ei, const float* __restrict__ ea,
                          float* __restrict__ deg, int E) {
  int e = blockIdx.x * blockDim.x + threadIdx.x;
  if (e < E) {
    int c = ei[E + e];                 // col = edge_index[1][e]
    atomicAdd(&deg[c], ea[e]);
  }
}

// ---- Kernel 2: dinv = rsqrt(deg + 1)  (self loop weight 1), in place ----
__global__ void deg_to_dinv(float* __restrict__ deg, int N) {
  int n = blockIdx.x * blockDim.x + threadIdx.x;
  if (n < N) {
    float d = deg[n] + 1.0f;
    deg[n] = (d > 0.0f) ? rsqrtf(d) : 0.0f;
  }
}

// ---- Kernel 3: H = X * W via V_WMMA_F32_16X16X4_F32 ----
// X: [rows,128] row-major. W: [128,128] row-major (k, n). H: [rows,128].
// Block = 256 threads = 8 wave32 waves; each wave computes a 16x128 slab.
// W is staged in LDS pre-swizzled so each lane's B operand {W[k'][col],W[k'+1][col]}
// is one aligned 8-byte ds_load_b64 (no repack movs before the WMMA).
__global__ void gemm_wmma_f32(const float* __restrict__ X, const float* __restrict__ W,
                              float* __restrict__ H, int rows) {
  __shared__ float ldsW[D_FEAT * D_FEAT];            // 64 KB, shared by 8 waves
  const int tid = threadIdx.x;
  // swizzled cooperative store: W[d][f] -> lds[(d>>2)*512 + ((d>>1)&1)*256 + f*2 + (d&1)]
  for (int i = tid; i < D_FEAT * D_FEAT; i += blockDim.x) {
    int d = i >> 7;
    int f = i & 127;
    ldsW[(d >> 2) * 512 + ((d >> 1) & 1) * 256 + f * 2 + (d & 1)] = W[i];
  }
  __syncthreads();

  const int wave = tid >> 5;
  const int lane = tid & 31;
  const int half = lane >> 4;          // 0: K {0,1}; 1: K {2,3}  (ISA A-matrix 16x4 layout)
  const int l16  = lane & 15;

  const int rowBase = (blockIdx.x * 8 + wave) * 16;
  int arow = rowBase + l16;
  if (arow > rows - 1) arow = rows - 1;               // clamp: keep EXEC all-1s for WMMA
  const float* xrow = X + (long)arow * D_FEAT;

  v8f acc[8] = {};

  // per-lane LDS base for B (in floats); advances 512 floats (2 KB) per k-step
  const float* bptr = ldsW + half * 256 + l16 * 2;

  for (int k = 0; k < D_FEAT; k += 4) {
    v2f a = *(const v2f*)(xrow + k + 2 * half);
#pragma unroll
    for (int ct = 0; ct < 8; ++ct) {
      v2f b = *(const v2f*)(bptr + ct * 32);          // ds_load_b64, imm offset 128*ct B
      acc[ct] = __builtin_amdgcn_wmma_f32_16x16x4_f32(false, a, false, b,
                                                      (short)0, acc[ct], false, false);
    }
    bptr += 512;
  }

  // C/D layout: VGPR v, lane L -> row M = v + 8*(L>=16), col = L%16
#pragma unroll
  for (int ct = 0; ct < 8; ++ct) {
#pragma unroll
    for (int v = 0; v < 8; ++v) {
      int gr = rowBase + v + 8 * half;
      if (gr < rows) H[(long)gr * D_FEAT + ct * 16 + l16] = acc[ct][v];
    }
  }
}

// ---- Kernel 4: gather-scale-scatter over edges (atomics resolve in L2) ----
__global__ void edge_scatter(const int* __restrict__ ei, const float* __restrict__ ea,
                             const float* __restrict__ dinv, const float* __restrict__ H,
                             float* __restrict__ AGG, int E, int N) {
  long t = (long)blockIdx.x * blockDim.x + threadIdx.x;
  long total = (long)E * (D_FEAT / 4);
  if (t >= total) return;
  int e = (int)(t >> 5);               // 32 chunks of 4 features
  int c = (int)(t & 31) * 4;
  int r  = ei[e];
  int cn = ei[E + e];
  float nrm = dinv[r] * ea[e] * dinv[cn];
#pragma unroll
  for (int b = 0; b < BATCH; ++b) {
    const float4 h4 = *(const float4*)(H + ((long)b * N + r) * D_FEAT + c);
    float* dst = AGG + ((long)b * N + cn) * D_FEAT + c;
    atomicAdd(dst + 0, h4.x * nrm);
    atomicAdd(dst + 1, h4.y * nrm);
    atomicAdd(dst + 2, h4.z * nrm);
    atomicAdd(dst + 3, h4.w * nrm);
  }
}

// ---- Kernel 5: out = relu(agg + h*dinv^2 (self loop) + bias) ----
__global__ void finish_relu(const float* __restrict__ AGG, const float* __restrict__ H,
                            const float* __restrict__ dinv, const float* __restrict__ bias,
                            float* __restrict__ OUT, int N) {
  long t = (long)blockIdx.x * blockDim.x + threadIdx.x;
  long total = (long)BATCH * N * (D_FEAT / 4);
  if (t >= total) return;
  int c = (int)(t & 31) * 4;
  long bn = t >> 5;
  int n = (int)(bn % N);
  float s = dinv[n]; s *= s;
  const float4 a4 = *(const float4*)(AGG + bn * D_FEAT + c);
  const float4 h4 = *(const float4*)(H + bn * D_FEAT + c);
  float4 o;
  o.x = fmaxf(a4.x + h4.x * s + bias[c + 0], 0.0f);
  o.y = fmaxf(a4.y + h4.y * s + bias[c + 1], 0.0f);
  o.z = fmaxf(a4.z + h4.z * s + bias[c + 2], 0.0f);
  o.w = fmaxf(a4.w + h4.w * s + bias[c + 3], 0.0f);
  *(float4*)(OUT + bn * D_FEAT + c) = o;
}

extern "C" void kernel_launch(void* const* d_in, const int* in_sizes, int n_in,
                              void* d_out, int out_size, void* d_ws, size_t ws_size,
                              hipStream_t stream) {
  const float* x  = (const float*)d_in[0];
  const int*   ei = (const int*)d_in[1];
  const float* ea = (const float*)d_in[2];
  const float* W1 = (const float*)d_in[3];
  const float* b1 = (const float*)d_in[4];
  const float* W2 = (const float*)d_in[5];
  const float* b2 = (const float*)d_in[6];
  float* out = (float*)d_out;

  const int N = in_sizes[0] / (BATCH * D_FEAT);
  const int E = in_sizes[2];
  const long feat_elems = (long)BATCH * N * D_FEAT;

  char* ws = (char*)d_ws;
  float* dinv = (float*)ws;
  size_t off = ((size_t)N * 4 + 255) & ~(size_t)255;
  float* hA  = (float*)(ws + off);
  float* agg = (float*)(ws + off + (size_t)feat_elems * 4);

  // degree + symmetric normalization
  hipMemsetAsync(dinv, 0, (size_t)N * 4, stream);
  deg_accum<<<(E + 255) / 256, 256, 0, stream>>>(ei, ea, dinv, E);
  deg_to_dinv<<<(N + 255) / 256, 256, 0, stream>>>(dinv, N);

  const int rows = BATCH * N;
  const int gemmBlocks = (rows + 127) / 128;
  const long scatterT = (long)E * (D_FEAT / 4);
  const long finishT  = feat_elems / 4;

  for (int layer = 0; layer < 2; ++layer) {
    const float* in  = (layer == 0) ? x : out;
    const float* W   = (layer == 0) ? W1 : W2;
    const float* bia = (layer == 0) ? b1 : b2;
    gemm_wmma_f32<<<gemmBlocks, 256, 0, stream>>>(in, W, hA, rows);
    hipMemsetAsync(agg, 0, (size_t)feat_elems * 4, stream);
    edge_scatter<<<(int)((scatterT + 255) / 256), 256, 0, stream>>>(ei, ea, dinv, hA, agg, E, N);
    finish_relu<<<(int)((finishT + 255) / 256), 256, 0, stream>>>(agg, hA, dinv, bia, out, N);
  }
}